// ROI_Pool_25366076850846
// MI455X (gfx1250) — compile-verified
//
#include <hip/hip_runtime.h>

// ---------------------------------------------------------------------------
// ROI max pooling on MI455X (gfx1250).
//   feature_map : (C=256, H=128, W=128) f32
//   rois        : (N, 4) i32  = (y, x, rH, rW)
//   out         : (N, C, PH=8, PW=8) f32
//
// One workgroup per (n, c); 4 wave32 waves, each wave owns h-bins ph = wid
// and wid+4. Per bin: Tensor Data Mover DMA of the bin's row block into LDS
// (tensor_load_to_lds + s_wait_tensorcnt), column maxes via ds_load_b128
// with 2x-unrolled dual accumulators, then an LDS transpose reduce:
// one ds_store_b128 of per-lane column maxes, 4 lanes per w-bin serially
// max a strided quarter of the bin, and a 2-step shfl_xor quad combine.
// ---------------------------------------------------------------------------

typedef unsigned int u32x4 __attribute__((ext_vector_type(4)));
typedef int          i32x8 __attribute__((ext_vector_type(8)));
typedef int          i32x4 __attribute__((ext_vector_type(4)));

#define C_   256
#define H_   128
#define W_   128
#define PH_  8
#define PW_  8
#define MAXROWS 17                 // max rows in one h-bin: ceil(128/8)+1
#define TILEF (MAXROWS * W_)       // floats per wave tile buffer (2176)

__device__ __forceinline__ float4 max4(float4 a, float4 b) {
    float4 r;
    r.x = fmaxf(a.x, b.x);
    r.y = fmaxf(a.y, b.y);
    r.z = fmaxf(a.z, b.z);
    r.w = fmaxf(a.w, b.w);
    return r;
}

__global__ __launch_bounds__(128)
void roi_pool_tdm_kernel(const float* __restrict__ fm,
                         const int*   __restrict__ rois,
                         float*       __restrict__ out) {
    __shared__ __align__(16) float tile[4 * TILEF];   // 34,816 bytes

    const int c    = blockIdx.x;
    const int n    = blockIdx.y;
    const int wid  = threadIdx.x >> 5;
    const int lane = threadIdx.x & 31;

    const int y  = rois[n * 4 + 0];
    const int x  = rois[n * 4 + 1];
    const int rH = rois[n * 4 + 2];
    const int rW = rois[n * 4 + 3];

    const int rWp = (rW + 3) & ~3;          // LDS row stride, 16B aligned

    float* tw = &tile[wid * TILEF];
    // Low 32 bits of the flat LDS address == workgroup-relative LDS offset.
    const unsigned lds_base = (unsigned)(unsigned long long)(void*)tw;

    const unsigned long long chan_base =
        (unsigned long long)fm +
        4ull * ((unsigned long long)c * (H_ * W_) + (unsigned long long)x);

    // ---- ROI-invariant descriptor words (hoisted) ----
    const unsigned g0w0 = 1u;                              // count=1, user mode
    const int g1w0 = (2 << 16);                            // data_size = 4B
    const int g1w1 = ((W_ - x) & 0xffff) << 16;            // tensor_dim0 = W-x
    const int g1w3 = (rWp & 0xffff) << 16;                 // tile_dim0 = rWp
    const int g1w5 = W_;                                   // tensor_dim0_stride

    const float ninf = -__builtin_inff();
    const int bin = lane >> 2;              // w-bin owned by this lane quad
    const int sub = lane & 3;               // position within the quad
    const int ws  = (bin * rW) >> 3;                       // floor(bin*rW/8)
    const int we  = ((bin + 1) * rW + 7) >> 3;             // ceil(..) <= rW

    for (int ph = wid; ph < PH_; ph += 4) {
        const int hstart = (ph * rH) >> 3;                 // floor(ph*rH/8)
        const int hend   = ((ph + 1) * rH + 7) >> 3;       // ceil((ph+1)*rH/8)
        const int rows   = hend - hstart;                  // <= 17

        const unsigned long long gaddr =
            chan_base + 4ull * (unsigned long long)(y + hstart) * W_;

        // ------------------ Tensor DMA descriptor (D#) ------------------
        u32x4 g0;
        g0[0] = g0w0;
        g0[1] = lds_base;                                  // LDS byte address
        g0[2] = (unsigned)(gaddr & 0xffffffffu);           // global_addr[31:0]
        g0[3] = (unsigned)((gaddr >> 32) & 0x1ffffffu)     // global_addr[56:32]
              | (2u << 30);                                // type = 2 (image)

        i32x8 g1;
        g1[0] = g1w0;
        g1[1] = g1w1;
        g1[2] = (rows & 0xffff) << 16;                     // tensor_dim1 = rows
        g1[3] = g1w3;
        g1[4] = rows & 0xffff;                             // tile_dim1 = rows
        g1[5] = g1w5;
        g1[6] = 0;
        g1[7] = 0;

        i32x4 g2 = {0, 0, 0, 0};
        i32x4 g3 = {0, 0, 0, 0};

        // Force descriptor words wave-uniform (SGPR operands for the TDM).
        #pragma unroll
        for (int i2 = 0; i2 < 4; ++i2)
            g0[i2] = (unsigned)__builtin_amdgcn_readfirstlane((int)g0[i2]);
        #pragma unroll
        for (int i2 = 0; i2 < 8; ++i2)
            g1[i2] = __builtin_amdgcn_readfirstlane(g1[i2]);

#if defined(__clang_major__) && (__clang_major__ >= 23)
        i32x8 g4 = {0, 0, 0, 0, 0, 0, 0, 0};
        __builtin_amdgcn_tensor_load_to_lds(g0, g1, g2, g3, g4, 0);
#else
        __builtin_amdgcn_tensor_load_to_lds(g0, g1, g2, g3, 0);
#endif
        __builtin_amdgcn_s_wait_tensorcnt(0);
        __asm__ volatile("" ::: "memory");

        // ---------------- Column maxes over the bin's rows ----------------
        float4 cmA = {ninf, ninf, ninf, ninf};
        float4 cmB = cmA;
        const int rowstride4 = rWp >> 2;                   // float4 per row
        if ((lane << 2) < rWp) {
            const float4* twv = (const float4*)tw;
            int r = 0;
            for (; r + 1 < rows; r += 2) {                 // 2x unroll, dual acc
                float4 v0 = twv[r * rowstride4 + lane];    // ds_load_b128
                float4 v1 = twv[(r + 1) * rowstride4 + lane];
                cmA = max4(cmA, v0);
                cmB = max4(cmB, v1);
            }
            if (r < rows) cmA = max4(cmA, twv[r * rowstride4 + lane]);
        }
        const float4 cm = max4(cmA, cmB);

        // ------- LDS transpose reduce: 4 lanes per w-bin, 2 shuffles -------
        ((float4*)tw)[lane] = cm;           // ds_store_b128; DS is in-order
        const float* sc = tw;
        float m = ninf;
        for (int w = ws + sub; w < we; w += 4)
            m = fmaxf(m, sc[w]);            // ds_load_b32 (lanes of this quad)
        m = fmaxf(m, __shfl_xor(m, 1, 32));
        m = fmaxf(m, __shfl_xor(m, 2, 32));

        if (sub == 0) {
            out[(((size_t)n * C_ + c) * PH_ + ph) * PW_ + bin] = m;
        }
        __asm__ volatile("" ::: "memory");  // keep next TDM after LDS reads
    }
}

extern "C" void kernel_launch(void* const* d_in, const int* in_sizes, int n_in,
                              void* d_out, int out_size, void* d_ws, size_t ws_size,
                              hipStream_t stream) {
    const float* fm   = (const float*)d_in[0];
    const int*   rois = (const int*)d_in[1];
    float*       out  = (float*)d_out;

    const int N = in_sizes[1] / 4;   // rois is (N, 4)

    dim3 grid(C_, N);
    dim3 block(128);
    roi_pool_tdm_kernel<<<grid, block, 0, stream>>>(fm, rois, out);

    (void)n_in; (void)out_size; (void)d_ws; (void)ws_size;
}